// OuterMean_1915555414566
// MI455X (gfx1250) — compile-verified
//
#include <hip/hip_runtime.h>

typedef __attribute__((ext_vector_type(2))) float v2f;
typedef __attribute__((ext_vector_type(4))) float v4f;
typedef __attribute__((ext_vector_type(8))) float v8f;

#define MDIM 256   // m (reduced away)
#define IDIM 512   // i == j
#define DDIM 128   // d
#define HDIM 128   // h

// ---------------------------------------------------------------------------
// Kernel 1: per-row LayerNorm of x (m,i,d) fused with mean over m.
// One block per i (512 blocks, 8 waves). Wave w handles m = 8t+w; lane l owns
// d = 4l..4l+3 (float4). Single-pass sum / sum-of-squares wave reduction,
// cross-wave combine through 4 KB LDS. Output: xn_mean (512 x 128).
// ---------------------------------------------------------------------------
__global__ __launch_bounds__(256) void ln_mean_kernel(
    const float* __restrict__ x, const float* __restrict__ g,
    const float* __restrict__ b, float* __restrict__ xn_mean)
{
    const int i    = blockIdx.x;
    const int lane = threadIdx.x & 31;
    const int wave = threadIdx.x >> 5;
    const int d0   = lane * 4;

    const v4f gv = *(const v4f*)(g + d0);
    const v4f bv = *(const v4f*)(b + d0);

    v4f acc = {0.f, 0.f, 0.f, 0.f};
    for (int t = 0; t < MDIM / 8; ++t) {
        const int m = t * 8 + wave;
        const v4f v = *(const v4f*)(x + ((size_t)m * IDIM + i) * DDIM + d0);
        float s1 = v.x + v.y + v.z + v.w;
        float s2 = v.x * v.x + v.y * v.y + v.z * v.z + v.w * v.w;
#pragma unroll
        for (int off = 16; off >= 1; off >>= 1) {
            s1 += __shfl_xor(s1, off, 32);
            s2 += __shfl_xor(s2, off, 32);
        }
        const float mu  = s1 * (1.0f / DDIM);
        const float var = s2 * (1.0f / DDIM) - mu * mu;
        const float rs  = rsqrtf(var + 1e-5f);
        const v4f dv = {v.x - mu, v.y - mu, v.z - mu, v.w - mu};
        acc.x += dv.x * rs * gv.x + bv.x;
        acc.y += dv.y * rs * gv.y + bv.y;
        acc.z += dv.z * rs * gv.z + bv.z;
        acc.w += dv.w * rs * gv.w + bv.w;
    }

    __shared__ float lds[8][DDIM];
    lds[wave][d0 + 0] = acc.x;
    lds[wave][d0 + 1] = acc.y;
    lds[wave][d0 + 2] = acc.z;
    lds[wave][d0 + 3] = acc.w;
    __syncthreads();

    if (threadIdx.x < 32) {
        v4f o = {0.f, 0.f, 0.f, 0.f};
#pragma unroll
        for (int w = 0; w < 8; ++w) {
            o.x += lds[w][d0 + 0];
            o.y += lds[w][d0 + 1];
            o.z += lds[w][d0 + 2];
            o.w += lds[w][d0 + 3];
        }
        const float sc = 1.0f / MDIM;
        o.x *= sc; o.y *= sc; o.z *= sc; o.w *= sc;
        *(v4f*)(xn_mean + (size_t)i * DDIM + d0) = o;
    }
}

// ---------------------------------------------------------------------------
// Kernel 2: fp32 WMMA GEMM  D = A(Mx128) * B(128x128) + bias, row-major.
// One wave per 16x16 D tile, K swept in steps of 4 via V_WMMA_F32_16X16X4_F32.
// Fragment layouts per CDNA5 ISA 7.12.2:
//   A 16x4: lane l -> M = l&15; half hi = l>>4 -> {K0,K1} (hi=0) / {K2,K3}
//           (hi=1), i.e. a contiguous float2 at A[m][k0+2*hi].
//   B 4x16: VGPR0 rows K0/K2 across lane halves, VGPR1 rows K1/K3.
//   C/D:    VGPR v, lane l -> D[v + 8*(l>>4)][l&15].
// ---------------------------------------------------------------------------
__global__ __launch_bounds__(32) void gemm_wmma_f32(
    const float* __restrict__ A, const float* __restrict__ B,
    const float* __restrict__ bias, float* __restrict__ D)
{
    const int tm   = blockIdx.x * 16;
    const int tn   = blockIdx.y * 16;
    const int lane = threadIdx.x;
    const int lo   = lane & 15;
    const int hi   = lane >> 4;

    const float* arow = A + (size_t)(tm + lo) * DDIM;

    v8f c = {0.f, 0.f, 0.f, 0.f, 0.f, 0.f, 0.f, 0.f};
#pragma unroll 4
    for (int k0 = 0; k0 < DDIM; k0 += 4) {
        const v2f a = *(const v2f*)(arow + k0 + 2 * hi);
        v2f bf;
        bf.x = B[(size_t)(k0 + 2 * hi + 0) * HDIM + tn + lo];
        bf.y = B[(size_t)(k0 + 2 * hi + 1) * HDIM + tn + lo];
        c = __builtin_amdgcn_wmma_f32_16x16x4_f32(
                /*neg_a=*/false, a, /*neg_b=*/false, bf,
                /*c_mod=*/(short)0, c, /*reuse_a=*/false, /*reuse_b=*/false);
    }

    const float bval = bias ? bias[tn + lo] : 0.0f;
#pragma unroll
    for (int v = 0; v < 8; ++v) {
        D[(size_t)(tm + v + 8 * hi) * HDIM + tn + lo] = c[v] + bval;
    }
}

// ---------------------------------------------------------------------------
// Kernel 3: streaming outer sum  out[i,j,:] = L[i,:] + R[j,:].
// This writes 134 MB and is the roofline cost (~5.8 us at 23.3 TB/s), so:
// each wave emits one full contiguous 512 B row (32 x float4), non-temporal
// stores (write-once data), L row held in registers, R rows (256 KB total)
// serve from L2. Grid: (i, j-chunk of 64), 256 threads = 8 j's per pass.
// ---------------------------------------------------------------------------
__global__ __launch_bounds__(256) void outer_sum_kernel(
    const v4f* __restrict__ L4, const v4f* __restrict__ R4,
    v4f* __restrict__ out4)
{
    const int i  = blockIdx.x;
    const int q  = threadIdx.x & 31;   // float4 index within the d-row
    const int jj = threadIdx.x >> 5;   // 0..7
    const v4f lv = L4[(size_t)i * 32 + q];
    const int jbase = blockIdx.y * 64;
#pragma unroll
    for (int t = 0; t < 8; ++t) {
        const int j = jbase + t * 8 + jj;
        const v4f rv = R4[(size_t)j * 32 + q];
        v4f o;
        o.x = lv.x + rv.x; o.y = lv.y + rv.y;
        o.z = lv.z + rv.z; o.w = lv.w + rv.w;
        __builtin_nontemporal_store(o, out4 + ((size_t)i * IDIM + j) * 32 + q);
    }
}

// ---------------------------------------------------------------------------
extern "C" void kernel_launch(void* const* d_in, const int* in_sizes, int n_in,
                              void* d_out, int out_size, void* d_ws, size_t ws_size,
                              hipStream_t stream) {
    const float* x       = (const float*)d_in[0];
    const float* ln_g    = (const float*)d_in[1];
    const float* ln_b    = (const float*)d_in[2];
    const float* w_left  = (const float*)d_in[3];
    const float* b_left  = (const float*)d_in[4];
    const float* w_right = (const float*)d_in[5];
    const float* b_right = (const float*)d_in[6];
    const float* w_out   = (const float*)d_in[7];
    const float* b_out   = (const float*)d_in[8];
    float* out = (float*)d_out;

    // Workspace layout: 5 x (512*128) fp32 = 1.25 MB
    const size_t MAT = (size_t)IDIM * HDIM;
    float* xn_mean = (float*)d_ws;
    float* t_left  = xn_mean + MAT;
    float* t_right = t_left + MAT;
    float* Lbuf    = t_right + MAT;
    float* Rbuf    = Lbuf + MAT;

    // 1) LayerNorm + mean over m  -> xn_mean (512x128)
    ln_mean_kernel<<<dim3(IDIM), dim3(256), 0, stream>>>(x, ln_g, ln_b, xn_mean);

    // 2) T_left = xn_mean @ w_left + b_left ; T_right likewise
    dim3 ggrid(IDIM / 16, HDIM / 16);
    gemm_wmma_f32<<<ggrid, dim3(32), 0, stream>>>(xn_mean, w_left,  b_left,  t_left);
    gemm_wmma_f32<<<ggrid, dim3(32), 0, stream>>>(xn_mean, w_right, b_right, t_right);

    // 3) L = T_left @ w_out + b_out ; R = T_right @ w_out  (b_out folded into L)
    gemm_wmma_f32<<<ggrid, dim3(32), 0, stream>>>(t_left,  w_out, b_out,  Lbuf);
    gemm_wmma_f32<<<ggrid, dim3(32), 0, stream>>>(t_right, w_out, nullptr, Rbuf);

    // 4) out[i,j,:] = L[i,:] + R[j,:]   (134 MB streaming write)
    outer_sum_kernel<<<dim3(IDIM, IDIM / 64), dim3(256), 0, stream>>>(
        (const v4f*)Lbuf, (const v4f*)Rbuf, (v4f*)out);
}